// SparseMoEPerTokenFFN_42743514530358
// MI455X (gfx1250) — compile-verified
//
#include <hip/hip_runtime.h>
#include <math.h>

// Problem constants (match reference)
#define Bb 8
#define Tt 64
#define Ee 4
#define Dd 256
#define Hh 1024

typedef __attribute__((ext_vector_type(2))) float v2f;
typedef __attribute__((ext_vector_type(8))) float v8f;

// padded LDS row strides (break 64-bank conflicts on stride-256/1024 rows)
#define LXS (Dd + 2)   // 258
#define LHS (Hh + 2)   // 1026

__global__ void zero_out_kernel(float* __restrict__ out, int n) {
    int i = blockIdx.x * blockDim.x + threadIdx.x;
    if (i < n) out[i] = 0.0f;
}

// gates[b,t,e] = relu(x[b,t,:] . Wr[t,e,:] + br[t,e]); loss += 0.01*g/(B*T)
__global__ void gates_kernel(const float* __restrict__ x,
                             const float* __restrict__ Wr,
                             const float* __restrict__ br,
                             float* __restrict__ gates,
                             float* __restrict__ loss) {
    int idx = blockIdx.x * blockDim.x + threadIdx.x;
    if (idx >= Bb * Tt * Ee) return;
    int e = idx % Ee;
    int t = (idx / Ee) % Tt;
    int b = idx / (Ee * Tt);
    const float* xv = x + ((size_t)b * Tt + t) * Dd;
    const float* wv = Wr + ((size_t)t * Ee + e) * Dd;
    float s = br[t * Ee + e];
#pragma unroll 8
    for (int d = 0; d < Dd; ++d) s += xv[d] * wv[d];
    float g = s > 0.0f ? s : 0.0f;
    gates[idx] = g;  // layout [b][t][e]
    atomicAdd(loss, 0.01f * g / (float)(Bb * Tt));
}

// One workgroup per (t,e): GEMM1 [8x256]x[256x1024] -> GELU -> LDS,
// GEMM2 [8x1024]x[1024x256] -> gate -> atomic accumulate into out[b,t,:].
__launch_bounds__(256)
__global__ void moe_ffn_kernel(const float* __restrict__ x,
                               const float* __restrict__ fc1,
                               const float* __restrict__ b1,
                               const float* __restrict__ fc2,
                               const float* __restrict__ b2,
                               const float* __restrict__ gates,
                               float* __restrict__ out) {
    __shared__ float ldsX[16 * LXS];   // X tile, rows 8..15 zero
    __shared__ float ldsH[16 * LHS];   // hidden tile, rows 8..15 zero

    const int t = blockIdx.x / Ee;
    const int e = blockIdx.x % Ee;
    const int tid  = threadIdx.x;
    const int lane = tid & 31;
    const int wave = tid >> 5;
    const int col  = lane & 15;            // N column / M row index per WMMA layout
    const int khalf = (lane >> 4) << 1;    // upper half-wave handles K+2

    // Stage X[t] into LDS: rows 0..7 = batch rows, rows 8..15 = 0
    for (int i = tid; i < 16 * Dd; i += 256) {
        int r = i >> 8;     // / 256
        int c = i & 255;
        float v = (r < Bb) ? x[((size_t)r * Tt + t) * Dd + c] : 0.0f;
        ldsX[r * LXS + c] = v;
    }
    __syncthreads();

    const size_t te = (size_t)t * Ee + e;
    const float* W1 = fc1 + te * (size_t)Dd * Hh;  // [D][H] row-major
    const float* W2 = fc2 + te * (size_t)Hh * Dd;  // [H][D] row-major
    const float* B1 = b1 + te * Hh;
    const float* B2 = b2 + te * Dd;

    // ---------------- GEMM1: Hid = gelu(X * W1 + b1) ----------------
    // 64 N-chunks of 16 cols; each wave owns 8 chunks.
    for (int c8 = 0; c8 < 8; ++c8) {
        const int n0 = (wave * 8 + c8) * 16;
        v8f acc = {};
        const float* wcol = W1 + n0 + col;
#pragma unroll 4
        for (int k = 0; k < Dd; k += 4) {
            const int kk = k + khalf;
            v2f a = *(const v2f*)(&ldsX[col * LXS + kk]);   // A: M=col, K=kk,kk+1
            v2f bv;
            bv.x = wcol[(size_t)kk * Hh];                    // B: K=kk,   N=n0+col
            bv.y = wcol[(size_t)(kk + 1) * Hh];              // B: K=kk+1, N=n0+col
            acc = __builtin_amdgcn_wmma_f32_16x16x4_f32(
                false, a, false, bv, (short)0, acc, false, false);
        }
        const float bias = B1[n0 + col];
        const bool lowhalf = (lane < 16);
#pragma unroll
        for (int i = 0; i < 8; ++i) {
            float v = acc[i] + bias;
            float gel = 0.5f * v * (1.0f + erff(v * 0.70710678118f)); // exact GELU
            int row = lowhalf ? i : (8 + i);       // upper half writes ghost rows
            ldsH[row * LHS + n0 + col] = lowhalf ? gel : 0.0f;
        }
    }
    __syncthreads();

    // ---------------- GEMM2: Out += gate * (Hid * W2 + b2) ----------------
    // 16 N-chunks of 16 cols; each wave owns 2 chunks.
    for (int c2 = 0; c2 < 2; ++c2) {
        const int n0 = (wave * 2 + c2) * 16;
        v8f acc = {};
        const float* wcol = W2 + n0 + col;
#pragma unroll 4
        for (int k = 0; k < Hh; k += 4) {
            const int kk = k + khalf;
            v2f a = *(const v2f*)(&ldsH[col * LHS + kk]);
            v2f bv;
            bv.x = wcol[(size_t)kk * Dd];
            bv.y = wcol[(size_t)(kk + 1) * Dd];
            acc = __builtin_amdgcn_wmma_f32_16x16x4_f32(
                false, a, false, bv, (short)0, acc, false, false);
        }
        if (lane < 16) {  // only M rows 0..7 (in VGPRs 0..7 of low half) are real
            const int n = n0 + col;
            const float bias2 = B2[n];
#pragma unroll
            for (int i = 0; i < 8; ++i) {
                float g = gates[((size_t)i * Tt + t) * Ee + e];   // b = i
                float val = (acc[i] + bias2) * g;
                atomicAdd(&out[((size_t)i * Tt + t) * Dd + n], val);
            }
        }
    }
}

extern "C" void kernel_launch(void* const* d_in, const int* in_sizes, int n_in,
                              void* d_out, int out_size, void* d_ws, size_t ws_size,
                              hipStream_t stream) {
    const float* x   = (const float*)d_in[0];  // [B,T,D]
    const float* fc1 = (const float*)d_in[1];  // [T,E,D,H]
    const float* b1  = (const float*)d_in[2];  // [T,E,H]
    const float* fc2 = (const float*)d_in[3];  // [T,E,H,D]
    const float* b2  = (const float*)d_in[4];  // [T,E,D]
    const float* Wr  = (const float*)d_in[5];  // [T,E,D]
    const float* br  = (const float*)d_in[6];  // [T,E]

    float* out   = (float*)d_out;              // [B,T,D] flat + loss scalar at end
    float* gates = (float*)d_ws;               // B*T*E floats of scratch

    const int n_out = Bb * Tt * Dd + 1;
    zero_out_kernel<<<(n_out + 255) / 256, 256, 0, stream>>>(out, n_out);

    const int n_gates = Bb * Tt * Ee;
    gates_kernel<<<(n_gates + 255) / 256, 256, 0, stream>>>(
        x, Wr, br, gates, out + (size_t)Bb * Tt * Dd);

    moe_ffn_kernel<<<Tt * Ee, 256, 0, stream>>>(x, fc1, b1, fc2, b2, gates, out);
}